// GraphFourierKANLayer_14293651161796
// MI455X (gfx1250) — compile-verified
//
#include <hip/hip_runtime.h>
#include <hip/hip_bf16.h>
#include <stdint.h>

// ---------------------------------------------------------------------------
// GraphFourierKAN layer for MI455X (gfx1250).
// Memory-bound on streaming fourier_coeffs (512 MB, reused only B=2 times):
// ~22 us at 23.3 TB/s.  Weights are DMA'd 8 KB/chunk into LDS with the
// Tensor Data Mover (tensor_load_to_lds, TENSORcnt), double-buffered and
// overlapped with V_WMMA_F32_16X16X4_F32 contraction (f32 in/out, no
// precision loss).  Compute (537 MFLOP + 4M v_sin/v_cos) is negligible.
// ---------------------------------------------------------------------------

typedef float v2f __attribute__((ext_vector_type(2)));
typedef float v8f __attribute__((ext_vector_type(8)));
typedef unsigned int v4u __attribute__((ext_vector_type(4)));
typedef int v4i __attribute__((ext_vector_type(4)));
typedef int v8i __attribute__((ext_vector_type(8)));

namespace cfg {
constexpr int B = 2;
constexpr int N = 2048;
constexpr int D = 64;
constexpr int G = 8;
constexpr int O = 64;
constexpr int E = N * 16;          // 32768 edges
constexpr int K = D * G;           // 512 contraction length per trig part
constexpr int KC = 32;             // k-rows per TDM chunk (32*64*4B = 8 KB)
constexpr int NCHUNK = K / KC;     // 16 chunks
constexpr long long PART_STRIDE = (long long)N * K * O;  // fc part stride
}

#if defined(__has_builtin)
#if __has_builtin(__builtin_amdgcn_tensor_load_to_lds) && \
    __has_builtin(__builtin_amdgcn_s_wait_tensorcnt)
#define USE_TDM 1
#endif
#endif
#ifndef USE_TDM
#define USE_TDM 0
#endif

// --- Kernel 1: zero the segment-sum workspace (deterministic each launch) ---
__global__ void zero_ws_kernel(float* ws, int n) {
    int t = blockIdx.x * blockDim.x + threadIdx.x;
    if (t < n) ws[t] = 0.0f;
}

// --- Kernel 2: edge aggregation: sums[b,dst,d] += x[b,src,d]; counts[dst]++ --
__global__ void aggregate_kernel(const float* __restrict__ x,
                                 const int* __restrict__ edge_index,
                                 float* __restrict__ sums,
                                 float* __restrict__ counts) {
    int t = blockIdx.x * blockDim.x + threadIdx.x;   // E*D threads
    int e = t >> 6;              // / D
    int d = t & (cfg::D - 1);
    if (e >= cfg::E) return;
    int src = edge_index[e];             // row 0 of (2,E)
    int dst = edge_index[cfg::E + e];    // row 1 of (2,E)
#pragma unroll
    for (int b = 0; b < cfg::B; ++b) {
        atomicAdd(&sums[((long long)b * cfg::N + dst) * cfg::D + d],
                  x[((long long)b * cfg::N + src) * cfg::D + d]);
    }
    if (d == 0) atomicAdd(&counts[dst], 1.0f);
}

#if USE_TDM
// Issue one TDM 2D-tile load: fc chunk (KC rows x 64 o, f32) -> LDS.
__device__ __forceinline__ void tdm_issue(const float* src, unsigned lds_off) {
    unsigned long long ga = (unsigned long long)(uintptr_t)src;
    v4u g0;
    g0.x = 1u;                                  // count=1, user desc, no gather
    g0.y = lds_off;                             // lds_addr (bytes)
    g0.z = (unsigned)(ga & 0xffffffffu);        // global_addr[31:0]
    g0.w = (unsigned)((ga >> 32) & 0x01ffffffu) // global_addr[56:32]
           | (2u << 30);                        // type = 2 ("image")
    v8i g1;
    g1[0] = 0x20000;                // workgroup_mask=0, data_size=2 (4B)
    g1[1] = cfg::O << 16;           // tensor_dim0[15:0] = 64 (bits 79:48)
    g1[2] = cfg::K << 16;           // tensor_dim1[15:0] = 512 (bits 111:80)
    g1[3] = cfg::O << 16;           // tile_dim0 = 64 (bits 127:112)
    g1[4] = cfg::KC;                // tile_dim1 = 32 (bits 143:128)
    g1[5] = cfg::O;                 // tensor_dim0_stride = 64 (bits 207:160)
    g1[6] = 0;                      // stride hi / tensor_dim1_stride lo
    g1[7] = 0;
    v4i gz = {0, 0, 0, 0};          // groups 2/3 unused (2D tensor)
#if __clang_major__ >= 23
    v8i gz8 = {0, 0, 0, 0, 0, 0, 0, 0};
    __builtin_amdgcn_tensor_load_to_lds(g0, g1, gz, gz, gz8, 0);
#else
    __builtin_amdgcn_tensor_load_to_lds(g0, g1, gz, gz, 0);
#endif
}
#endif

// --- Kernel 3: per-node Fourier-KAN contraction via TDM + WMMA --------------
// Block = 1 node, 256 threads = 8 waves.
// wave w: p = w>>2 (0=cos,1=sin), o-tile = w&3 (16 outputs each).
// Waves 0 and 4 drive the TDM weight stream (8 KB chunks, double-buffered).
// Per wave per chunk: 8 x V_WMMA_F32_16X16X4_F32 over 32 k-rows.
//   A (16x4 f32): rows M=0,1 = activations act[p][b][k], rows 2..15 zero;
//                 lanes 0-15 hold K=4t+{0,1}, lanes 16-31 hold K=4t+{2,3}.
//   B (4x16 f32): weight tile W[k, obase+0..15] from LDS chunk buffer.
//   C/D (16x16):  acc[0] lanes0-15 = y[b=0,o], acc[1] lanes0-15 = y[b=1,o].
__global__ __launch_bounds__(256)
void fourier_kan_kernel(const float* __restrict__ fc,     // (2,N,D,G,O)
                        const float* __restrict__ bias,   // (1,N,O)
                        const float* __restrict__ sums,   // (B,N,D)
                        const float* __restrict__ counts, // (N)
                        float* __restrict__ out) {        // (B,N,O)
    __shared__ float xhat[cfg::B][cfg::D];              // 0.5 KB
    __shared__ float act[2][cfg::B][cfg::K];            // 8 KB trig table
    __shared__ float ypart[2][cfg::B][cfg::O];          // 1 KB partials
#if USE_TDM
    __shared__ float wbuf[2][2][cfg::KC * cfg::O];      // 32 KB [buf][part]
#endif

    const int n   = blockIdx.x;
    const int tid = threadIdx.x;
    const int wave  = tid >> 5;
    const int lane  = tid & 31;
    const int p     = wave >> 2;           // 0 = cos part, 1 = sin part
    const int obase = (wave & 3) << 4;     // o-tile base
    const int m     = lane & 15;           // output column index
    const int khalf = (lane >> 4) << 1;    // 0 for lanes 0-15, 2 for 16-31
    const float amask = (m < cfg::B) ? 1.0f : 0.0f;   // zero A rows M>=2
    const int arow = m & 1;                // safe LDS row for masked lanes

    const float* wpart = fc + (long long)p * cfg::PART_STRIDE
                            + (long long)n * (cfg::K * cfg::O);

#if USE_TDM
    // Prologue: DMA chunk 0 of each trig part into buffer 0.
    if (wave == 0) tdm_issue(wpart, (unsigned)(uintptr_t)&wbuf[0][0][0]);
    if (wave == 4) tdm_issue(wpart, (unsigned)(uintptr_t)&wbuf[0][1][0]);
#endif

    // x_hat = segment mean (overlaps with TDM chunk 0 in flight).
    const float inv_cnt = 1.0f / fmaxf(counts[n], 1.0f);
    if (tid < cfg::B * cfg::D) {
        int b = tid >> 6, d = tid & (cfg::D - 1);
        xhat[b][d] = sums[((long long)b * cfg::N + n) * cfg::D + d] * inv_cnt;
    }
    __syncthreads();

    // Activation table: act[p][b][d*G+g] = trig((g+1) * xhat[b][d])
    for (int i = tid; i < 2 * cfg::B * cfg::K; i += 256) {
        int pp  = i >> 10;
        int rem = i & 1023;
        int b   = rem >> 9;
        int k   = rem & (cfg::K - 1);
        int d   = k >> 3;            // G = 8
        int g   = k & (cfg::G - 1);
        float arg = (float)(g + 1) * xhat[b][d];
        act[pp][b][k] = pp ? __sinf(arg) : __cosf(arg);
    }

    v8f acc = {};

#if USE_TDM
    for (int ic = 0; ic < cfg::NCHUNK; ++ic) {
        const int buf = ic & 1;
        if (wave == 0 || wave == 4)
            __builtin_amdgcn_s_wait_tensorcnt(0);    // chunk ic landed in LDS
        __syncthreads();   // chunk ic visible; compute of ic-1 finished
        if (ic + 1 < cfg::NCHUNK) {                  // DMA chunk ic+1 now
            const float* nxt = wpart + (ic + 1) * (cfg::KC * cfg::O);
            if (wave == 0)
                tdm_issue(nxt, (unsigned)(uintptr_t)&wbuf[buf ^ 1][0][0]);
            if (wave == 4)
                tdm_issue(nxt, (unsigned)(uintptr_t)&wbuf[buf ^ 1][1][0]);
        }
        const float* wb = &wbuf[buf][p][khalf * cfg::O + obase + m];
#pragma unroll
        for (int j = 0; j < cfg::KC / 4; ++j) {      // 8 WMMA per chunk
            const int kb = ic * cfg::KC + j * 4 + khalf;
            v2f a;
            a.x = amask * act[p][arow][kb];
            a.y = amask * act[p][arow][kb + 1];
            v2f bm;
            bm.x = wb[j * 4 * cfg::O];               // LDS row kb
            bm.y = wb[j * 4 * cfg::O + cfg::O];      // LDS row kb+1
            // 8 args: (neg_a, A, neg_b, B, c_mod, C, reuse_a, reuse_b)
            acc = __builtin_amdgcn_wmma_f32_16x16x4_f32(
                false, a, false, bm, (short)0, acc, false, false);
        }
        __syncthreads();   // all waves done reading wbuf[buf]
    }
#else
    // Fallback: direct global weight stream.
    __syncthreads();
    const float* wp = wpart + (long long)khalf * cfg::O + obase + m;
#pragma unroll 4
    for (int kt = 0; kt < cfg::K / 4; ++kt) {
        if ((kt & 15) == 0 && kt + 16 < cfg::K / 4)
            __builtin_prefetch(wp + (kt + 16) * (4 * cfg::O), 0, 1);
        const int kb = kt * 4 + khalf;
        v2f a;
        a.x = amask * act[p][arow][kb];
        a.y = amask * act[p][arow][kb + 1];
        const float* w = wp + kt * (4 * cfg::O);
        v2f bm;
        bm.x = w[0];
        bm.y = w[cfg::O];
        acc = __builtin_amdgcn_wmma_f32_16x16x4_f32(
            false, a, false, bm, (short)0, acc, false, false);
    }
#endif

    // acc[0] lanes0-15 -> y[b=0, obase+lane]; acc[1] lanes0-15 -> y[b=1,...]
    if (lane < 16) {
        ypart[p][0][obase + lane] = acc[0];
        ypart[p][1][obase + lane] = acc[1];
    }
    __syncthreads();

    // Combine cos+sin partials, add bias, store.
    if (tid < cfg::B * cfg::O) {
        int b = tid >> 6, o = tid & (cfg::O - 1);
        out[((long long)b * cfg::N + n) * cfg::O + o] =
            ypart[0][b][o] + ypart[1][b][o] + bias[(long long)n * cfg::O + o];
    }
}

// ---------------------------------------------------------------------------
extern "C" void kernel_launch(void* const* d_in, const int* in_sizes, int n_in,
                              void* d_out, int out_size, void* d_ws, size_t ws_size,
                              hipStream_t stream) {
    const float* x      = (const float*)d_in[0];   // (B,N,D) f32
    const int*   ei     = (const int*)d_in[1];     // (2,E) int
    const float* fc     = (const float*)d_in[2];   // (2,N,D,G,O) f32
    const float* bias   = (const float*)d_in[3];   // (1,N,O) f32
    float*       out    = (float*)d_out;           // (B,N,O) f32

    float* sums   = (float*)d_ws;                            // B*N*D f32
    float* counts = sums + (size_t)cfg::B * cfg::N * cfg::D; // N f32

    const int ws_elems = cfg::B * cfg::N * cfg::D + cfg::N;
    zero_ws_kernel<<<(ws_elems + 255) / 256, 256, 0, stream>>>(sums, ws_elems);

    const int agg_threads = cfg::E * cfg::D;
    aggregate_kernel<<<agg_threads / 256, 256, 0, stream>>>(x, ei, sums, counts);

    fourier_kan_kernel<<<cfg::N, 256, 0, stream>>>(fc, bias, sums, counts, out);
}